// Fundamental2Pose_28132035789013
// MI455X (gfx1250) — compile-verified
//
#include <hip/hip_runtime.h>
#include <stdint.h>

// ---------------------------------------------------------------------------
// Problem constants / workspace layout (floats)
// ---------------------------------------------------------------------------
#define NBATCH 32
#define TBL_PER_B 60                               // 12 (Kinv rows, padded) + 4*12 (P2=[R|t] rows)
#define TBL_FLOATS (NBATCH * TBL_PER_B)            // 1920 floats = 7680 B
#define WS_SCORES_OFF TBL_FLOATS                   // 32*4 floats
#define WS_COUNTS_OFF (TBL_FLOATS + NBATCH * 4)    // 32 floats

typedef float v2f  __attribute__((ext_vector_type(2)));
typedef float v8f  __attribute__((ext_vector_type(8)));
typedef unsigned int u32x4 __attribute__((ext_vector_type(4)));
typedef int  i32x4 __attribute__((ext_vector_type(4)));
typedef int  i32x8 __attribute__((ext_vector_type(8)));

// ---------------------------------------------------------------------------
// Fully-unrolled cyclic Jacobi eigensolver for small symmetric matrices.
// ---------------------------------------------------------------------------
template <int NDIM, int SWEEPS>
__device__ __forceinline__ void jacobi_eig(float G[NDIM][NDIM], float V[NDIM][NDIM]) {
#pragma unroll
  for (int i = 0; i < NDIM; ++i)
#pragma unroll
    for (int j = 0; j < NDIM; ++j) V[i][j] = (i == j) ? 1.0f : 0.0f;

#pragma unroll
  for (int sw = 0; sw < SWEEPS; ++sw) {
#pragma unroll
    for (int p = 0; p < NDIM - 1; ++p) {
#pragma unroll
      for (int q = p + 1; q < NDIM; ++q) {
        float apq   = G[p][q];
        float theta = (G[q][q] - G[p][p]) / (2.0f * apq);   // inf if apq==0 -> t==0
        float t     = 1.0f / (fabsf(theta) + sqrtf(theta * theta + 1.0f));
        t           = (theta < 0.0f) ? -t : t;
        t           = (fabsf(apq) < 1e-20f) ? 0.0f : t;     // NaN-safe select
        float c     = 1.0f / sqrtf(t * t + 1.0f);
        float sn    = t * c;
#pragma unroll
        for (int r = 0; r < NDIM; ++r) {
          float grp = G[r][p], grq = G[r][q];
          G[r][p] = c * grp - sn * grq;
          G[r][q] = sn * grp + c * grq;
        }
#pragma unroll
        for (int r = 0; r < NDIM; ++r) {
          float gpr = G[p][r], gqr = G[q][r];
          G[p][r] = c * gpr - sn * gqr;
          G[q][r] = sn * gpr + c * gqr;
          float vrp = V[r][p], vrq = V[r][q];
          V[r][p] = c * vrp - sn * vrq;
          V[r][q] = sn * vrp + c * vrq;
        }
      }
    }
  }
}

__device__ __forceinline__ float col3(const float V[3][3], int r, int c) {
  return (c == 0) ? V[r][0] : ((c == 1) ? V[r][1] : V[r][2]);
}

// ---------------------------------------------------------------------------
// Kernel A: per-batch pose candidates. 1 block x 32 lanes (one batch/lane).
// ---------------------------------------------------------------------------
__global__ __launch_bounds__(32) void pose_kernel(const float* __restrict__ Fg,
                                                  const float* __restrict__ Kg,
                                                  float* __restrict__ ws) {
  const int b = threadIdx.x;
  if (b >= NBATCH) return;

  float F[3][3], K[3][3];
#pragma unroll
  for (int i = 0; i < 3; ++i)
#pragma unroll
    for (int j = 0; j < 3; ++j) {
      F[i][j] = Fg[b * 9 + i * 3 + j];
      K[i][j] = Kg[b * 9 + i * 3 + j];
    }

  // E = K^T F K
  float T[3][3], E[3][3];
#pragma unroll
  for (int i = 0; i < 3; ++i)
#pragma unroll
    for (int j = 0; j < 3; ++j)
      T[i][j] = K[0][i] * F[0][j] + K[1][i] * F[1][j] + K[2][i] * F[2][j];
#pragma unroll
  for (int i = 0; i < 3; ++i)
#pragma unroll
    for (int j = 0; j < 3; ++j)
      E[i][j] = T[i][0] * K[0][j] + T[i][1] * K[1][j] + T[i][2] * K[2][j];

  // V from eigen(E^T E); U from E*v / sigma; sigma3 ~ 0 => u2 = u0 x u1.
  float G[3][3];
#pragma unroll
  for (int i = 0; i < 3; ++i)
#pragma unroll
    for (int j = 0; j < 3; ++j)
      G[i][j] = E[0][i] * E[0][j] + E[1][i] * E[1][j] + E[2][i] * E[2][j];

  float V[3][3];
  jacobi_eig<3, 6>(G, V);

  float ev[3] = {G[0][0], G[1][1], G[2][2]};
  int idx[3]  = {0, 1, 2};
#define SW_(a, b_)                                                 \
  if (ev[b_] > ev[a]) {                                            \
    float te = ev[a]; ev[a] = ev[b_]; ev[b_] = te;                 \
    int ti = idx[a]; idx[a] = idx[b_]; idx[b_] = ti;               \
  }
  SW_(0, 1) SW_(0, 2) SW_(1, 2)
#undef SW_

  float v0[3], v1[3], v2[3];
#pragma unroll
  for (int r = 0; r < 3; ++r) {
    v0[r] = col3(V, r, idx[0]);
    v1[r] = col3(V, r, idx[1]);
    v2[r] = col3(V, r, idx[2]);
  }

  float u0[3], u1[3], u2[3];
#pragma unroll
  for (int r = 0; r < 3; ++r)
    u0[r] = E[r][0] * v0[0] + E[r][1] * v0[1] + E[r][2] * v0[2];
  float n0 = 1.0f / sqrtf(u0[0] * u0[0] + u0[1] * u0[1] + u0[2] * u0[2] + 1e-30f);
#pragma unroll
  for (int r = 0; r < 3; ++r) u0[r] *= n0;
#pragma unroll
  for (int r = 0; r < 3; ++r)
    u1[r] = E[r][0] * v1[0] + E[r][1] * v1[1] + E[r][2] * v1[2];
  float d01 = u0[0] * u1[0] + u0[1] * u1[1] + u0[2] * u1[2];
#pragma unroll
  for (int r = 0; r < 3; ++r) u1[r] -= d01 * u0[r];
  float n1 = 1.0f / sqrtf(u1[0] * u1[0] + u1[1] * u1[1] + u1[2] * u1[2] + 1e-30f);
#pragma unroll
  for (int r = 0; r < 3; ++r) u1[r] *= n1;
  u2[0] = u0[1] * u1[2] - u0[2] * u1[1];
  u2[1] = u0[2] * u1[0] - u0[0] * u1[2];
  u2[2] = u0[0] * u1[1] - u0[1] * u1[0];

  float cx = v1[1] * v2[2] - v1[2] * v2[1];
  float cy = v1[2] * v2[0] - v1[0] * v2[2];
  float cz = v1[0] * v2[1] - v1[1] * v2[0];
  float detV = v0[0] * cx + v0[1] * cy + v0[2] * cz;
  float sv = (detV < 0.0f) ? -1.0f : 1.0f;
#pragma unroll
  for (int r = 0; r < 3; ++r) { v0[r] *= sv; v1[r] *= sv; v2[r] *= sv; }

  // R1 = U*W*V^T, R2 = U*W^T*V^T ; (U*W) cols = [u1,-u0,u2], (U*W^T) = [-u1,u0,u2]
  float R1[3][3], R2[3][3];
#pragma unroll
  for (int i = 0; i < 3; ++i)
#pragma unroll
    for (int j = 0; j < 3; ++j) {
      R1[i][j] =  u1[i] * v0[j] - u0[i] * v1[j] + u2[i] * v2[j];
      R2[i][j] = -u1[i] * v0[j] + u0[i] * v1[j] + u2[i] * v2[j];
    }

  // Kinv via adjugate
  float c00 = K[1][1] * K[2][2] - K[1][2] * K[2][1];
  float c01 = -(K[1][0] * K[2][2] - K[1][2] * K[2][0]);
  float c02 = K[1][0] * K[2][1] - K[1][1] * K[2][0];
  float id = 1.0f / (K[0][0] * c00 + K[0][1] * c01 + K[0][2] * c02);
  float inv[3][3];
  inv[0][0] =  (K[1][1] * K[2][2] - K[1][2] * K[2][1]) * id;
  inv[0][1] = -(K[0][1] * K[2][2] - K[0][2] * K[2][1]) * id;
  inv[0][2] =  (K[0][1] * K[1][2] - K[0][2] * K[1][1]) * id;
  inv[1][0] = -(K[1][0] * K[2][2] - K[1][2] * K[2][0]) * id;
  inv[1][1] =  (K[0][0] * K[2][2] - K[0][2] * K[2][0]) * id;
  inv[1][2] = -(K[0][0] * K[1][2] - K[0][2] * K[1][0]) * id;
  inv[2][0] =  (K[1][0] * K[2][1] - K[1][1] * K[2][0]) * id;
  inv[2][1] = -(K[0][0] * K[2][1] - K[0][1] * K[2][0]) * id;
  inv[2][2] =  (K[0][0] * K[1][1] - K[0][1] * K[1][0]) * id;

  float* tb = ws + b * TBL_PER_B;
#pragma unroll
  for (int i = 0; i < 3; ++i) {
    tb[i * 4 + 0] = inv[i][0];
    tb[i * 4 + 1] = inv[i][1];
    tb[i * 4 + 2] = inv[i][2];
    tb[i * 4 + 3] = 0.0f;
  }
#pragma unroll
  for (int c = 0; c < 4; ++c) {
    float* P = tb + 12 + c * 12;
    float sg = (c & 1) ? -1.0f : 1.0f;
#pragma unroll
    for (int i = 0; i < 3; ++i) {
#pragma unroll
      for (int j = 0; j < 3; ++j) P[i * 4 + j] = (c < 2) ? R1[i][j] : R2[i][j];
      P[i * 4 + 3] = sg * u2[i];  // t = U[:,2]
    }
  }
}

// ---------------------------------------------------------------------------
__global__ void zero_kernel(float* __restrict__ ws) {
  int i = blockIdx.x * blockDim.x + threadIdx.x;
  if (i < NBATCH * 4 + NBATCH) ws[WS_SCORES_OFF + i] = 0.0f;
}

// ---------------------------------------------------------------------------
// Kernel B: cheirality test + WMMA-accumulator segment sum (branchless feed).
// Packed slot word: bits[7:0]=batch, bits[11:8]=cand flags, bit[12]=valid.
// ---------------------------------------------------------------------------
__global__ __launch_bounds__(256) void cheirality_kernel(
    const float2* __restrict__ pts1, const float2* __restrict__ pts2,
    const int* __restrict__ bidx, const float* __restrict__ ws_tbl,
    float* __restrict__ scores, float* __restrict__ counts, int n) {
  __shared__ float sh_tbl[TBL_FLOATS];          // declared FIRST -> LDS offset 0 (TDM dest)
  __shared__ float sh_acc[NBATCH * 4 + NBATCH]; // block-level partials (160 floats)

  for (int i = threadIdx.x; i < NBATCH * 4 + NBATCH; i += blockDim.x) sh_acc[i] = 0.0f;

#if defined(__HIP_DEVICE_COMPILE__) && __has_builtin(__builtin_amdgcn_tensor_load_to_lds) && \
    __has_builtin(__builtin_amdgcn_s_wait_tensorcnt)
  if (threadIdx.x == 0) {  // TDM issues per-wave regardless of EXEC; only wave 0 reaches here
    unsigned long long ga = (unsigned long long)(uintptr_t)ws_tbl;
    u32x4 g0;
    g0.x = 1u;                                               // count=1, user-mode D#
    g0.y = 0u;                                               // lds_addr = 0 (sh_tbl at LDS base)
    g0.z = (unsigned)(ga & 0xFFFFFFFFu);                     // global_addr[31:0]
    g0.w = (unsigned)((ga >> 32) & 0x1FFFFFFu) | (2u << 30); // global_addr[56:32] | type=2
    i32x8 g1;
    g1[0] = (2 << 16);                                        // wg_mask=0, data_size=2 (4B)
    g1[1] = (int)((TBL_FLOATS & 0xFFFF) << 16);               // tensor_dim0[15:0]
    g1[2] = (int)(((TBL_FLOATS >> 16) & 0xFFFF) | (1 << 16)); // tensor_dim0 hi | tensor_dim1=1
    g1[3] = (int)(TBL_FLOATS << 16);                          // tile_dim0 = 1920
    g1[4] = 1;                                                // tile_dim1 = 1, tile_dim2 = 0
    g1[5] = TBL_FLOATS;                                       // tensor_dim0_stride[31:0]
    g1[6] = 0;
    g1[7] = 0;
    i32x4 gz4 = {0, 0, 0, 0};
    i32x8 gz8 = {0, 0, 0, 0, 0, 0, 0, 0};
    __builtin_amdgcn_tensor_load_to_lds(g0, g1, gz4, gz4, gz8, 0);
    __builtin_amdgcn_s_wait_tensorcnt(0);
  }
#else
  for (int i = threadIdx.x; i < TBL_FLOATS; i += blockDim.x) sh_tbl[i] = ws_tbl[i];
#endif
  __syncthreads();

  const int lane    = threadIdx.x & 31;
  const int wave    = blockIdx.x * (blockDim.x >> 5) + (threadIdx.x >> 5);
  const int nwaves  = gridDim.x * (blockDim.x >> 5);
  const int nchunks = (n + 31) >> 5;
  const int m_row   = lane & 15;           // WMMA A row / B column owned by this lane
  const int k_base  = (lane < 16) ? 0 : 2; // WMMA K slots owned by this lane half

  // Loop-invariant column-class masks for the B operand (branchless feed).
  const int cls_flag_lo = (m_row < 4) ? 1 : 0;                 // scores, batch < 16
  const int cls_flag_hi = (m_row >= 4 && m_row < 8) ? 1 : 0;   // scores, batch >= 16
  const int cls_cnt_lo  = (m_row == 8) ? 1 : 0;                // counts, batch < 16
  const int cls_cnt_hi  = (m_row == 9) ? 1 : 0;                // counts, batch >= 16
  const int cbit        = m_row & 3;                           // candidate bit for flag cols

  v8f acc = {};  // running binned segment-sums (scores + counts)

  for (int ch = wave; ch < nchunks; ch += nwaves) {
    int i = ch * 32 + lane;
    int valid = (i < n) ? 1 : 0;
    int ic = valid ? i : (n - 1);  // clamp -> unconditional loads, no divergence

    // Speculative prefetch of next chunk (dropped by HW if untranslatable).
    int nexti = (ch + nwaves) * 32 + lane;
    nexti = (nexti < n) ? nexti : (n - 1);
    __builtin_prefetch((const void*)(pts1 + nexti), 0, 1);
    __builtin_prefetch((const void*)(pts2 + nexti), 0, 1);
    __builtin_prefetch((const void*)(bidx + nexti), 0, 1);

    int pb = bidx[ic];
    float2 q1 = pts1[ic];
    float2 q2 = pts2[ic];
    const float* tb = &sh_tbl[pb * TBL_PER_B];
    float p1x = tb[0] * q1.x + tb[1] * q1.y + tb[2];
    float p1y = tb[4] * q1.x + tb[5] * q1.y + tb[6];
    float p2x = tb[0] * q2.x + tb[1] * q2.y + tb[2];
    float p2y = tb[4] * q2.x + tb[5] * q2.y + tb[6];
    int fmask = 0;
#pragma unroll
    for (int c = 0; c < 4; ++c) {
      const float* P = tb + 12 + c * 12;  // rows of [R|t]
      float r0[4] = {-1.0f, 0.0f, p1x, 0.0f};
      float r1[4] = {0.0f, -1.0f, p1y, 0.0f};
      float r2[4], r3[4];
#pragma unroll
      for (int j = 0; j < 4; ++j) {
        r2[j] = p2x * P[8 + j] - P[j];
        r3[j] = p2y * P[8 + j] - P[4 + j];
      }
      float G[4][4];
#pragma unroll
      for (int a = 0; a < 4; ++a)
#pragma unroll
        for (int bb = 0; bb < 4; ++bb)
          G[a][bb] = r0[a] * r0[bb] + r1[a] * r1[bb] + r2[a] * r2[bb] + r3[a] * r3[bb];
      float V[4][4];
      jacobi_eig<4, 3>(G, V);
      int mi = 0; float mv = G[0][0];
      if (G[1][1] < mv) { mv = G[1][1]; mi = 1; }
      if (G[2][2] < mv) { mv = G[2][2]; mi = 2; }
      if (G[3][3] < mv) { mv = G[3][3]; mi = 3; }
      float X[4];
#pragma unroll
      for (int j = 0; j < 4; ++j)
        X[j] = (mi == 0) ? V[j][0] : (mi == 1) ? V[j][1] : (mi == 2) ? V[j][2] : V[j][3];
      float z1s = X[2] * X[3];  // sign(z1) ; z2*X3 below
      float z2s = (P[8] * X[0] + P[9] * X[1] + P[10] * X[2] + P[11] * X[3]) * X[3];
      int infront = ((z1s > 0.0f) & (z2s > 0.0f)) ? 1 : 0;
      fmask |= infront << c;
    }

    // --- WMMA binned accumulation: 8 groups of 4 point-slots, branchless ---
    int code = (pb | (fmask << 8) | (1 << 12)) * valid;  // packed slot word
#pragma unroll
    for (int g = 0; g < 8; ++g) {
      int s0 = g * 4 + k_base;
      int c0 = __shfl(code, s0, 32);
      int c1 = __shfl(code, s0 + 1, 32);

      int b0 = c0 & 0xFF, v0_ = (c0 >> 12) & 1, h0 = (b0 >= 16) ? 1 : 0;
      int b1 = c1 & 0xFF, v1_ = (c1 >> 12) & 1, h1 = (b1 >= 16) ? 1 : 0;
      int f0 = (c0 >> (8 + cbit)) & 1;
      int f1 = (c1 >> (8 + cbit)) & 1;

      int a0 = v0_ & (((b0 & 15) == m_row) ? 1 : 0);
      int a1 = v1_ & (((b1 & 15) == m_row) ? 1 : 0);
      int s0v = v0_ & ((cls_flag_lo & (h0 ^ 1) & f0) | (cls_flag_hi & h0 & f0) |
                       (cls_cnt_lo & (h0 ^ 1)) | (cls_cnt_hi & h0));
      int s1v = v1_ & ((cls_flag_lo & (h1 ^ 1) & f1) | (cls_flag_hi & h1 & f1) |
                       (cls_cnt_lo & (h1 ^ 1)) | (cls_cnt_hi & h1));

      v2f av, bv;
      av.x = (float)a0;  av.y = (float)a1;
      bv.x = (float)s0v; bv.y = (float)s1v;
      acc = __builtin_amdgcn_wmma_f32_16x16x4_f32(false, av, false, bv,
                                                  (short)0, acc, false, false);
    }
  }

  // Drain accumulator: lane owns column n=lane&15, rows m = r + (lane<16?0:8)
  {
    int n_col  = lane & 15;
    int m_base = (lane < 16) ? 0 : 8;
#pragma unroll
    for (int r = 0; r < 8; ++r) {
      float v = acc[r];
      int m = m_base + r;
      if (n_col < 4)       atomicAdd(&sh_acc[m * 4 + n_col], v);
      else if (n_col < 8)  atomicAdd(&sh_acc[(m + 16) * 4 + (n_col - 4)], v);
      else if (n_col == 8) atomicAdd(&sh_acc[NBATCH * 4 + m], v);
      else if (n_col == 9) atomicAdd(&sh_acc[NBATCH * 4 + m + 16], v);
    }
  }
  __syncthreads();
  for (int i = threadIdx.x; i < NBATCH * 4 + NBATCH; i += blockDim.x) {
    float v = sh_acc[i];
    if (v != 0.0f) {
      if (i < NBATCH * 4) atomicAdd(&scores[i], v);
      else                atomicAdd(&counts[i - NBATCH * 4], v);
    }
  }
}

// ---------------------------------------------------------------------------
// Kernel C: argmax + gather. out = [best_R(32*9) | best_t(32*3) | best_cand(32)]
// ---------------------------------------------------------------------------
__global__ __launch_bounds__(32) void finalize_kernel(const float* __restrict__ ws,
                                                      float* __restrict__ out) {
  const int b = threadIdx.x;
  if (b >= NBATCH) return;
  const float* sc = ws + WS_SCORES_OFF + b * 4;
  float cnt = ws[WS_COUNTS_OFF + b];
  int best = 0; float bs = sc[0];
#pragma unroll
  for (int c = 1; c < 4; ++c)
    if (sc[c] > bs) { bs = sc[c]; best = c; }  // strict '>' == first max (jnp.argmax)
  float has = (cnt > 0.0f) ? 1.0f : 0.0f;
  const float* P = ws + b * TBL_PER_B + 12 + best * 12;
#pragma unroll
  for (int i = 0; i < 3; ++i) {
#pragma unroll
    for (int j = 0; j < 3; ++j) out[b * 9 + i * 3 + j] = P[i * 4 + j] * has;
    out[NBATCH * 9 + b * 3 + i] = P[i * 4 + 3] * has;
  }
  out[NBATCH * 12 + b] = (cnt > 0.0f) ? (float)best : 0.0f;
}

// ---------------------------------------------------------------------------
extern "C" void kernel_launch(void* const* d_in, const int* in_sizes, int n_in,
                              void* d_out, int out_size, void* d_ws, size_t ws_size,
                              hipStream_t stream) {
  const float* F   = (const float*)d_in[0];
  const float* K   = (const float*)d_in[1];
  const float2* p1 = (const float2*)d_in[2];
  const float2* p2 = (const float2*)d_in[3];
  const int* bi    = (const int*)d_in[4];
  const int n      = in_sizes[4];
  float* ws = (float*)d_ws;

  zero_kernel<<<1, 256, 0, stream>>>(ws);
  pose_kernel<<<1, 32, 0, stream>>>(F, K, ws);
  cheirality_kernel<<<128, 256, 0, stream>>>(p1, p2, bi, ws,
                                             ws + WS_SCORES_OFF, ws + WS_COUNTS_OFF, n);
  finalize_kernel<<<1, 32, 0, stream>>>(ws, (float*)d_out);
}